// GlobalAttention_hype_52432960749619
// MI455X (gfx1250) — compile-verified
//
#include <hip/hip_runtime.h>
#include <math.h>

#define BB 8
#define TT 2048
#define SS 2048
#define DD 128
#define LDST 132   // padded LDS row stride (floats): 128 data + 4 pad (TDM pad feature)

typedef float v2f __attribute__((ext_vector_type(2)));
typedef float v4f __attribute__((ext_vector_type(4)));
typedef float v8f __attribute__((ext_vector_type(8)));
typedef unsigned int v4u __attribute__((ext_vector_type(4)));
typedef int v4i __attribute__((ext_vector_type(4)));
typedef int v8i __attribute__((ext_vector_type(8)));

__device__ __forceinline__ float artanh_c(float x) {
    x = fminf(fmaxf(x, -1.f + 1e-7f), 1.f - 1e-7f);
    return 0.5f * __logf((1.f + x) / (1.f - x));
}

// Tensor Data Mover: 2D tile load, rows x 128 f32, row stride 128 f32 in memory,
// LDS rows padded to 132 f32 via D# pad (interval 128 DW -> code 6, amount 4 DW -> code 3).
__device__ __forceinline__ void tdm_load_tile(unsigned lds_addr, const void* gaddr,
                                              unsigned tensor_rows) {
    unsigned long long ga = (unsigned long long)(uintptr_t)gaddr;
    v4u g0;
    g0.x = 1u;                                     // count=1, user descriptor
    g0.y = lds_addr;                               // LDS byte address
    g0.z = (unsigned)ga;                           // global addr [31:0]
    g0.w = (unsigned)(ga >> 32) | (2u << 30);      // global addr [56:32] | type=2
    v8i g1;
    g1[0] = (int)((2u << 16) | (1u << 20) | (6u << 22) | (3u << 25)); // 4B elems, pad on
    g1[1] = (int)(128u << 16);                     // tensor_dim0 = 128 (bits 79:48)
    g1[2] = (int)(tensor_rows << 16);              // dim0 hi | tensor_dim1 low16 (bits 111:80)
    g1[3] = (int)(128u << 16);                     // tensor_dim1 hi | tile_dim0 = 128
    g1[4] = 16;                                    // tile_dim1 = 16, tile_dim2 = 0
    g1[5] = 128;                                   // tensor_dim0_stride = 128 elems
    g1[6] = 0;
    g1[7] = 0;
    v4i z4 = {0, 0, 0, 0};
    v8i z8 = {0, 0, 0, 0, 0, 0, 0, 0};
    __builtin_amdgcn_tensor_load_to_lds(g0, g1, z4, z4, z8, 0);
}

// ---------------------------------------------------------------- kernel 0 --
// y2[b,s] = ||mem[b,s]||^2 ; gamma[b,s] = 2/max(1-y2,1e-15)
__global__ void k_rownorm(const float* __restrict__ memb,
                          float* __restrict__ y2w, float* __restrict__ gw) {
    int wave = threadIdx.x >> 5, lane = threadIdx.x & 31;
    int row = blockIdx.x * 8 + wave;              // 0 .. B*S-1
    const float* p = memb + (size_t)row * DD;
    float s = 0.f;
#pragma unroll
    for (int u = 0; u < 4; ++u) { float v = p[lane + 32 * u]; s += v * v; }
    for (int m = 16; m >= 1; m >>= 1) s += __shfl_xor(s, m, 32);
    if (lane == 0) { y2w[row] = s; gw[row] = 2.f / fmaxf(1.f - s, 1e-15f); }
}

// ---------------------------------------------------------------- kernel A --
// fused: dist GEMM (wmma f32 16x16x4, hs tiles via TDM) -> hyperbolic score ->
// mask -> raw scores to out[t,b,s]; online softmax stats (M,L) to workspace.
__global__ void k_score(const float* __restrict__ src, const float* __restrict__ memb,
                        const float* __restrict__ y2w,
                        const float* __restrict__ betaw, const float* __restrict__ betab,
                        const int* __restrict__ mlen,
                        float* __restrict__ alignBuf,
                        float* __restrict__ Mw, float* __restrict__ Lw) {
    __shared__ __align__(16) float hsT[16 * LDST];
    __shared__ float y2t[16];
    __shared__ float x2w[64];

    int b  = blockIdx.x >> 5;                // 32 t-blocks per batch
    int tb = blockIdx.x & 31;
    int wave = threadIdx.x >> 5, lane = threadIdx.x & 31;
    int t0 = tb * 64 + wave * 16;
    int li = lane & 15, hi = lane >> 4;
    unsigned hs_lds = (unsigned)(uintptr_t)&hsT[0];

    float w  = betaw[0], bbv = betab[0];
    float aw = fabsf(w), sw = (w >= 0.f) ? 1.f : -1.f;
    int ml = mlen[b];

    // A fragments: 16 t-rows x 128 d, ISA A-layout (lanes 0-15: K=k0,k0+1;
    // lanes 16-31: K=k0+2,k0+3)
    v2f afr[32];
    {
        const float* rp = src + ((size_t)b * TT + t0 + li) * DD + 2 * hi;
#pragma unroll
        for (int ks = 0; ks < 32; ++ks) afr[ks] = *(const v2f*)(rp + 4 * ks);
    }
    // x2 per t-row: partial per lane, combine across halves
    float x2p = 0.f;
#pragma unroll
    for (int ks = 0; ks < 32; ++ks) x2p += afr[ks].x * afr[ks].x + afr[ks].y * afr[ks].y;
    x2p += __shfl_xor(x2p, 16, 32);
    if (hi == 0) x2w[wave * 16 + li] = x2p;
    __syncthreads();

    float mrun[8], lrun[8];
#pragma unroll
    for (int i = 0; i < 8; ++i) { mrun[i] = -__builtin_inff(); lrun[i] = 0.f; }

    for (int s0 = 0; s0 < SS; s0 += 16) {
        __syncthreads();            // previous tile fully consumed
        if (wave == 0) {            // TDM DMA: 16x128 hs tile -> padded LDS rows
            tdm_load_tile(hs_lds, memb + ((size_t)b * SS + s0) * DD, (unsigned)(SS - s0));
            __builtin_amdgcn_s_wait_tensorcnt(0);
        }
        if (threadIdx.x < 16) y2t[threadIdx.x] = y2w[b * SS + s0 + threadIdx.x];
        __syncthreads();            // tile visible to all waves

        v8f acc = {};
#pragma unroll
        for (int ks = 0; ks < 32; ++ks) {
            int kk = 4 * ks + 2 * hi;
            v2f bfr = *(const v2f*)(hsT + li * LDST + kk);
            acc = __builtin_amdgcn_wmma_f32_16x16x4_f32(false, afr[ks], false, bfr,
                                                        (short)0, acc, false, false);
        }

        float y2v = y2t[li];
        int s = s0 + li;
        bool valid = (s < ml);
#pragma unroll
        for (int i = 0; i < 8; ++i) {
            int M = i + 8 * hi;
            float x2v = x2w[wave * 16 + M];
            float xy = acc[i];
            float diff2 = fmaxf(x2v + y2v - 2.f * xy, 0.f);
            float dn = fmaxf(1.f - 2.f * xy + x2v * y2v, 1e-15f);
            float dist = 2.f * artanh_c(__fsqrt_rn(diff2 / dn));
            // 1-D mobius linear with (beta_w, beta_b)
            float mm = sw * tanhf(aw * artanh_c(dist));
            float num = (1.f + 2.f * mm * bbv + bbv * bbv) * mm + (1.f - mm * mm) * bbv;
            float dd2 = fmaxf(1.f + 2.f * mm * bbv + mm * mm * bbv * bbv, 1e-15f);
            float r = num / dd2;
            if (fabsf(r) > 0.996f) r = copysignf(0.996f, r);   // project, 1-4e-3
            float a = valid ? -r : -__builtin_inff();
            int t = t0 + M;
            alignBuf[((size_t)t * BB + b) * SS + s] = a;       // raw score, (t,b,s)
            if (valid) {
                float mo = mrun[i];
                if (a > mo) { lrun[i] *= __expf(mo - a); mrun[i] = a; mo = a; }
                lrun[i] += __expf(a - mo);
            }
        }
    }
    // combine online stats across the 16 lanes of each half-wave
#pragma unroll
    for (int i = 0; i < 8; ++i) {
        float m = mrun[i], l = lrun[i];
#pragma unroll
        for (int d = 1; d < 16; d <<= 1) {
            float om = __shfl_xor(m, d, 32);
            float ol = __shfl_xor(l, d, 32);
            float nm = fmaxf(m, om);
            float c1 = (l  > 0.f) ? l  * __expf(m  - nm) : 0.f;
            float c2 = (ol > 0.f) ? ol * __expf(om - nm) : 0.f;
            m = nm; l = c1 + c2;
        }
        if (l <= 0.f) { l = 1.f; m = 0.f; }
        if (li == i) {
            int t = t0 + i + 8 * hi;
            Mw[b * TT + t] = m;
            Lw[b * TT + t] = l;
        }
    }
}

// ---------------------------------------------------------------- kernel B --
// normalize scores in place (final align output) and wmma-accumulate
// nom = (P*gamma) @ mem  (gamma folded into P fragments), den = P@(gamma-1).
// mem tiles streamed by TDM; nom parked in attn_h output region.
__global__ void k_attnv(const float* __restrict__ memb, const float* __restrict__ gw,
                        const float* __restrict__ Mw, const float* __restrict__ Lw,
                        float* __restrict__ alignBuf, float* __restrict__ nomBuf,
                        float* __restrict__ denW) {
    __shared__ __align__(16) float vg[16 * LDST];
    __shared__ float gt[16];

    int b  = blockIdx.x >> 5;
    int tb = blockIdx.x & 31;
    int wave = threadIdx.x >> 5, lane = threadIdx.x & 31;
    int t0 = tb * 64 + wave * 16;
    int li = lane & 15, hi = lane >> 4;
    int trow = t0 + li;
    unsigned vg_lds = (unsigned)(uintptr_t)&vg[0];

    float mrow = Mw[b * TT + trow];
    float linv = 1.f / Lw[b * TT + trow];

    v8f zero = {};
    v8f acc[8];
#pragma unroll
    for (int j = 0; j < 8; ++j) acc[j] = zero;
    float dacc = 0.f;

    float* arow = alignBuf + ((size_t)trow * BB + b) * SS;

    for (int s0 = 0; s0 < SS; s0 += 16) {
        __syncthreads();
        if (wave == 0) {            // TDM DMA: raw 16x128 mem tile -> padded LDS
            tdm_load_tile(vg_lds, memb + ((size_t)b * SS + s0) * DD, (unsigned)(SS - s0));
            __builtin_amdgcn_s_wait_tensorcnt(0);
        }
        if (threadIdx.x < 16) gt[threadIdx.x] = gw[b * SS + s0 + threadIdx.x];
        __syncthreads();

        // P fragments in A-layout, gamma folded in; each tile element touched once
        v2f pfr[4];
#pragma unroll
        for (int ks = 0; ks < 4; ++ks) {
            int kk = 4 * ks + 2 * hi;
            v2f raw = *(const v2f*)(arow + s0 + kk);
            v2f p;
            p.x = __expf(raw.x - mrow) * linv;
            p.y = __expf(raw.y - mrow) * linv;
            *(v2f*)(arow + s0 + kk) = p;                 // final align_vectors
            v2f pg;
            pg.x = p.x * gt[kk];
            pg.y = p.y * gt[kk + 1];
            dacc += (pg.x - p.x) + (pg.y - p.y);         // p*(gamma-1)
            pfr[ks] = pg;
        }
#pragma unroll
        for (int j = 0; j < 8; ++j) {
#pragma unroll
            for (int ks = 0; ks < 4; ++ks) {
                int kk = 4 * ks + 2 * hi;
                v2f bfr;
                bfr.x = vg[kk * LDST + 16 * j + li];
                bfr.y = vg[(kk + 1) * LDST + 16 * j + li];
                acc[j] = __builtin_amdgcn_wmma_f32_16x16x4_f32(false, pfr[ks], false, bfr,
                                                               (short)0, acc[j], false, false);
            }
        }
    }
    // nom -> attn_h region (overwritten by k_tail), den -> workspace
#pragma unroll
    for (int j = 0; j < 8; ++j) {
#pragma unroll
        for (int i = 0; i < 8; ++i) {
            int t = t0 + i + 8 * hi;
            int d = 16 * j + li;
            nomBuf[((size_t)t * BB + b) * DD + d] = acc[j][i];
        }
    }
    dacc += __shfl_xor(dacc, 16, 32);
    if (hi == 0) denW[b * TT + t0 + li] = dacc;
}

// ---------------------------------------------------------------- kernel C --
// tail: mobius_scalar_mul(0.5, nom/den), concat with source, mobius linear
// with W_out/b_out, project; one wave per (b,t) row.
__global__ void k_tail(const float* __restrict__ src, const float* __restrict__ Wout,
                       const float* __restrict__ bout, const float* __restrict__ denW,
                       float* __restrict__ attnBuf) {
    __shared__ float xs[8][256];
    int wave = threadIdx.x >> 5, lane = threadIdx.x & 31;
    int idx = blockIdx.x * 8 + wave;              // b*T + t
    int b = idx / TT, t = idx % TT;

    float den = denW[idx];
    den = (den >= 0.f) ? fmaxf(den, 1e-10f) : fminf(den, -1e-10f);

    float* nom = attnBuf + ((size_t)t * BB + b) * DD;
    float v[4]; float tn2 = 0.f;
#pragma unroll
    for (int u = 0; u < 4; ++u) { v[u] = nom[lane + 32 * u] / den; tn2 += v[u] * v[u]; }
    for (int m = 16; m >= 1; m >>= 1) tn2 += __shfl_xor(tn2, m, 32);
    float tn = __fsqrt_rn(tn2);
    float cf = tanhf(0.5f * artanh_c(tn)) / fmaxf(tn, 1e-15f);

    float sv[4]; float xn2 = 0.f;
#pragma unroll
    for (int u = 0; u < 4; ++u) {
        v[u] *= cf;
        sv[u] = src[((size_t)b * TT + t) * DD + lane + 32 * u];
        xn2 += v[u] * v[u] + sv[u] * sv[u];
        xs[wave][lane + 32 * u] = v[u];
        xs[wave][128 + lane + 32 * u] = sv[u];
    }
    for (int m = 16; m >= 1; m >>= 1) xn2 += __shfl_xor(xn2, m, 32);
    float xn = __fsqrt_rn(xn2);
    __syncthreads();

    float mx[4]; float mxn2 = 0.f;
#pragma unroll
    for (int u = 0; u < 4; ++u) {
        int j = lane + 32 * u;
        const float* wr = Wout + (size_t)j * 256;
        float a0 = 0.f;
        for (int k = 0; k < 256; k += 4) {
            v4f w4 = *(const v4f*)(wr + k);
            a0 += w4.x * xs[wave][k] + w4.y * xs[wave][k + 1]
                + w4.z * xs[wave][k + 2] + w4.w * xs[wave][k + 3];
        }
        mx[u] = a0; mxn2 += a0 * a0;
    }
    for (int m = 16; m >= 1; m >>= 1) mxn2 += __shfl_xor(mxn2, m, 32);
    float mxn = __fsqrt_rn(mxn2);
    float fac = (mxn > 0.f) ? tanhf((mxn / fmaxf(xn, 1e-15f)) * artanh_c(xn)) / mxn : 0.f;

    float r[4], bv[4]; float xy = 0.f, y2 = 0.f;
#pragma unroll
    for (int u = 0; u < 4; ++u) {
        r[u] = fac * mx[u];
        bv[u] = bout[lane + 32 * u];
        xy += r[u] * bv[u]; y2 += bv[u] * bv[u];
    }
    for (int m = 16; m >= 1; m >>= 1) { xy += __shfl_xor(xy, m, 32); y2 += __shfl_xor(y2, m, 32); }
    float x2 = fac * fac * mxn2;
    float c1 = 1.f + 2.f * xy + y2;
    float c2 = 1.f - x2;
    float dn = fmaxf(1.f + 2.f * xy + x2 * y2, 1e-15f);

    float h[4]; float hn2 = 0.f;
#pragma unroll
    for (int u = 0; u < 4; ++u) { h[u] = (c1 * r[u] + c2 * bv[u]) / dn; hn2 += h[u] * h[u]; }
    for (int m = 16; m >= 1; m >>= 1) hn2 += __shfl_xor(hn2, m, 32);
    float hn = __fsqrt_rn(hn2);
    float sc = (hn > 0.996f) ? 0.996f / fmaxf(hn, 1e-15f) : 1.f;
#pragma unroll
    for (int u = 0; u < 4; ++u) nom[lane + 32 * u] = h[u] * sc;    // final attn_h
}

// -------------------------------------------------------------------- launch
extern "C" void kernel_launch(void* const* d_in, const int* in_sizes, int n_in,
                              void* d_out, int out_size, void* d_ws, size_t ws_size,
                              hipStream_t stream) {
    (void)in_sizes; (void)n_in; (void)out_size; (void)ws_size;
    const float* src   = (const float*)d_in[0];   // (B,T,D)
    const float* memb  = (const float*)d_in[1];   // (B,S,D)
    const float* Wout  = (const float*)d_in[2];   // (D,2D)
    const float* bout  = (const float*)d_in[3];   // (D)
    const float* betaw = (const float*)d_in[4];   // (1,1)
    const float* betab = (const float*)d_in[5];   // (1)
    const int*   mlen  = (const int*)d_in[6];     // (B)

    float* attnBuf  = (float*)d_out;                           // (T,B,D)
    float* alignBuf = attnBuf + (size_t)TT * BB * DD;          // (T,B,S)

    float* ws   = (float*)d_ws;
    float* y2w  = ws;                                          // B*S
    float* gw   = ws + (size_t)BB * SS;                        // B*S
    float* Mw   = ws + (size_t)2 * BB * SS;                    // B*T
    float* Lw   = ws + (size_t)2 * BB * SS + BB * TT;          // B*T
    float* denW = ws + (size_t)2 * BB * SS + 2 * BB * TT;      // B*T

    k_rownorm<<<BB * SS / 8, 256, 0, stream>>>(memb, y2w, gw);
    k_score <<<BB * TT / 64, 128, 0, stream>>>(src, memb, y2w, betaw, betab, mlen,
                                               alignBuf, Mw, Lw);
    k_attnv <<<BB * TT / 64, 128, 0, stream>>>(memb, gw, Mw, Lw, alignBuf, attnBuf, denW);
    k_tail  <<<BB * TT / 8, 256, 0, stream>>>(src, Wout, bout, denW, attnBuf);
}